// Conv_Mixing_85641647882250
// MI455X (gfx1250) — compile-verified
//
#include <hip/hip_runtime.h>

#define BATCH 32
#define CC 384
#define CH 96
#define HW 1024
#define BC (BATCH*CC)                 // 12288
#define NELEM ((size_t)BATCH*CC*HW)   // 12582912 floats per branch tensor
#define KT 32
#define CPAD 36                       // comb LDS row stride (16B-aligned rows)

typedef __attribute__((ext_vector_type(2))) float v2f;
typedef __attribute__((ext_vector_type(8))) float v8f;

__device__ __forceinline__ float gelu_erf(float x) {
    return 0.5f * x * (1.0f + erff(x * 0.70710678118654752f));
}

// ---------------- Branch MLP: per-pixel 96 -> 12 (gelu) -> 96 -----------------
// grid = (B*4 pixel tiles, 2 mlp-chunks), block = 256 (one pixel per thread)
__global__ __launch_bounds__(256) void mlp_kernel(
    const float* __restrict__ x, const float* __restrict__ mw1, const float* __restrict__ mb1,
    const float* __restrict__ mw2, const float* __restrict__ mb2, float* __restrict__ out)
{
    __shared__ float sW1[CH*12];
    __shared__ float sW2[12*CH];
    __shared__ float sB1[12];
    __shared__ float sB2[CH];
    const int tid = threadIdx.x;
    const int b   = blockIdx.x >> 2;
    const int pix = ((blockIdx.x & 3) << 8) + tid;
    const int mi  = blockIdx.y;          // 0 -> chunk 0, 1 -> chunk 2
    const int cBase = mi * 192;

    for (int i = tid; i < CH*12; i += 256) sW1[i] = mw1[mi*CH*12 + i];
    for (int i = tid; i < 12*CH; i += 256) sW2[i] = mw2[mi*12*CH + i];
    if (tid < 12) sB1[tid] = mb1[mi*12 + tid];
    if (tid < CH) sB2[tid] = mb2[mi*CH + tid];
    __syncthreads();

    float h[12];
#pragma unroll
    for (int j = 0; j < 12; ++j) h[j] = sB1[j];
    const float* xp = x + (size_t)(b*CC + cBase) * HW + pix;
    for (int c = 0; c < CH; ++c) {
        float v = xp[c * HW];
#pragma unroll
        for (int j = 0; j < 12; ++j) h[j] += v * sW1[c*12 + j];
    }
#pragma unroll
    for (int j = 0; j < 12; ++j) h[j] = gelu_erf(h[j]);

    float* op = out + (size_t)(b*CC + cBase) * HW + pix;
    for (int c = 0; c < CH; ++c) {
        float s = sB2[c];
#pragma unroll
        for (int j = 0; j < 12; ++j) s += h[j] * sW2[j*CH + c];
        op[c * HW] = s;
    }
}

// ---------------- Branch depthwise conv (k = 1/3/5, pad = 0/1/2) --------------
// grid = (B*4 pixel tiles, 192 channels), block = 256
__global__ __launch_bounds__(256) void dwconv_kernel(
    const float* __restrict__ x, const float* __restrict__ cw, const float* __restrict__ cb,
    int k, int pad, float* __restrict__ out)
{
    const int tid = threadIdx.x;
    const int b   = blockIdx.x >> 2;
    const int pix = ((blockIdx.x & 3) << 8) + tid;
    const int ch  = blockIdx.y;          // 0..191
    const int ci  = ch / CH;             // weight set (chunk 1 -> 0, chunk 3 -> 1)
    const int cl  = ch % CH;
    const int cg  = (1 + 2*ci) * CH + cl;
    const int h = pix >> 5, w = pix & 31;

    const float* xp = x + (size_t)(b*CC + cg) * HW;
    const float* wp = cw + (ci*CH + cl) * k * k;
    float s = cb[ci*CH + cl];
    for (int kh = 0; kh < k; ++kh) {
        int ih = h - pad + kh;
        if ((unsigned)ih >= 32u) continue;
        for (int kw = 0; kw < k; ++kw) {
            int iw = w - pad + kw;
            if ((unsigned)iw >= 32u) continue;
            s += xp[ih*32 + iw] * wp[kh*k + kw];
        }
    }
    out[(size_t)(b*CC + cg) * HW + pix] = s;
}

// ---------------- Global mean of (x1+x2+x3) over H,W --------------------------
__global__ __launch_bounds__(256) void mean_kernel(
    const float* __restrict__ x1, const float* __restrict__ x2, const float* __restrict__ x3,
    float* __restrict__ a)
{
    __shared__ float sred[256];
    const size_t base = (size_t)blockIdx.x * HW;
    const int tid = threadIdx.x;
    float s = 0.f;
    for (int i = tid; i < HW; i += 256) s += x1[base+i] + x2[base+i] + x3[base+i];
    sred[tid] = s; __syncthreads();
    for (int off = 128; off > 0; off >>= 1) {
        if (tid < off) sred[tid] += sred[tid+off];
        __syncthreads();
    }
    if (tid == 0) a[blockIdx.x] = sred[0] * (1.0f/1024.0f);
}

// ---------------- Fusion MLP + softmax gate weights ---------------------------
// grid = B, block = 384
__global__ __launch_bounds__(384) void fusion_kernel(
    const float* __restrict__ a, const float* __restrict__ w1, const float* __restrict__ b1,
    const float* __restrict__ w2, const float* __restrict__ b2, float* __restrict__ wgt)
{
    __shared__ float sA[CC];
    __shared__ float sH[96];
    __shared__ float sG[3*CC];
    const int b = blockIdx.x, tid = threadIdx.x;
    sA[tid] = a[b*CC + tid];
    __syncthreads();
    if (tid < 96) {
        float s = b1[tid];
        for (int c = 0; c < CC; ++c) s += sA[c] * w1[c*96 + tid];
        sH[tid] = gelu_erf(s);
    }
    __syncthreads();
    for (int r = 0; r < 3; ++r) {
        int o = r*CC + tid;                       // covers all 1152 outputs
        float s = b2[o];
        for (int j = 0; j < 96; ++j) s += sH[j] * w2[j*(3*CC) + o];
        sG[o] = s;
    }
    __syncthreads();
    {   // g.reshape(B,C,3): g[b,c,r] = row[c*3+r]; softmax over r
        const int c = tid;
        float g0 = sG[c*3+0], g1 = sG[c*3+1], g2 = sG[c*3+2];
        float m = fmaxf(g0, fmaxf(g1, g2));
        float e0 = __expf(g0-m), e1 = __expf(g1-m), e2 = __expf(g2-m);
        float inv = 1.0f/(e0+e1+e2);
        wgt[0*BC + b*CC + c] = e0*inv;
        wgt[1*BC + b*CC + c] = e1*inv;
        wgt[2*BC + b*CC + c] = e2*inv;
    }
}

// ---------------- Fused combine + projection GEMM (WMMA f32 16x16x4) ----------
// out_b(384 x 1024) = P^T(384x384) @ comb_b(384x1024), comb fused from x1..x3.
// grid = B * 32 pixel-tiles = 1024 blocks, block = 256 (8 waves).
// Wave w owns M rows [48w, 48w+48) as 3x16 tiles, x 2 N tiles of 16 pixels.
// P^T K-slab is staged global->LDS via async DMA (ASYNCcnt) overlapping with
// the VALU computation of the gated comb tile.
__global__ __launch_bounds__(256) void proj_wmma_kernel(
    const float* __restrict__ x1, const float* __restrict__ x2, const float* __restrict__ x3,
    const float* __restrict__ wgt, const float* __restrict__ P, const float* __restrict__ pb,
    float* __restrict__ out)
{
    __shared__ float sP[KT*CC];                  // 48 KB: sP[kl*384 + d]
    __shared__ __align__(16) float sCm[KT*CPAD]; // comb tile [kl][p], padded rows

    const int tid  = threadIdx.x;
    const int lane = tid & 31;
    const int wave = tid >> 5;
    const int b     = blockIdx.x >> 5;
    const int pBase = (blockIdx.x & 31) << 5;
    const int mWave = wave * 48;
    const int nl = lane & 15;
    const int hi = lane >> 4;     // half-wave: K offset +2 per WMMA A/B layout

    const unsigned sPbase = (unsigned)(uintptr_t)&sP[0];   // LDS byte offset (low 32 bits)

    v8f acc[3][2] = {};

    for (int k0 = 0; k0 < CC; k0 += KT) {
        // ---- async DMA: contiguous 48KB P^T slab, global -> LDS (no VGPR round trip)
        {
            const float* gP = P + (size_t)k0 * CC;     // slab is contiguous
#pragma unroll
            for (int t = 0; t < (KT*CC)/(4*256); ++t) {   // 12 x b128 per thread
                const int idx16 = t*256 + tid;
                unsigned laddr = sPbase + (unsigned)(idx16 * 16);
                unsigned long long gaddr = (unsigned long long)(gP + idx16*4);
                asm volatile("global_load_async_to_lds_b128 %0, %1, off"
                             :: "v"(laddr), "v"(gaddr) : "memory");
            }
        }
        // ---- meanwhile: compute fused comb tile with VALU (w0*x1 + w1*x2 + w2*x3)
        {
            const int i = tid;                 // KT*8 float4 elements == 256 threads
            const int kl = i >> 3, p4 = i & 7;
            const int bcid = b*CC + (k0 + kl);
            const size_t g = (size_t)bcid * HW + pBase + p4*4;
            const float w0 = wgt[bcid], w1 = wgt[BC + bcid], w2 = wgt[2*BC + bcid];
            const float4 a1 = *(const float4*)(x1 + g);
            const float4 a2 = *(const float4*)(x2 + g);
            const float4 a3 = *(const float4*)(x3 + g);
            float4 cv;
            cv.x = w0*a1.x + w1*a2.x + w2*a3.x;
            cv.y = w0*a1.y + w1*a2.y + w2*a3.y;
            cv.z = w0*a1.z + w1*a2.z + w2*a3.z;
            cv.w = w0*a1.w + w1*a2.w + w2*a3.w;
            *(float4*)&sCm[kl*CPAD + p4*4] = cv;   // 16B-aligned (CPAD*4 % 16 == 0)
        }
        asm volatile("s_wait_asynccnt 0x0" ::: "memory");
        __syncthreads();

        for (int kk = 0; kk < KT; kk += 4) {
            const int kSel = kk + (hi << 1);   // lanes 0-15: K0/K1, lanes 16-31: K2/K3
            v2f bf0, bf1;
            bf0.x = sCm[kSel*CPAD + nl];       bf0.y = sCm[(kSel+1)*CPAD + nl];
            bf1.x = sCm[kSel*CPAD + 16 + nl];  bf1.y = sCm[(kSel+1)*CPAD + 16 + nl];
#pragma unroll
            for (int mt = 0; mt < 3; ++mt) {
                const int d = mWave + mt*16 + nl;
                v2f af;
                af.x = sP[kSel*CC + d];
                af.y = sP[(kSel+1)*CC + d];
                acc[mt][0] = __builtin_amdgcn_wmma_f32_16x16x4_f32(
                    false, af, false, bf0, (short)0, acc[mt][0], false, false);
                acc[mt][1] = __builtin_amdgcn_wmma_f32_16x16x4_f32(
                    false, af, false, bf1, (short)0, acc[mt][1], false, false);
            }
        }
        __syncthreads();
    }

    // C/D layout: VGPR r -> M=r (lanes 0-15) / M=r+8 (lanes 16-31), N = lane&15
#pragma unroll
    for (int mt = 0; mt < 3; ++mt) {
#pragma unroll
        for (int nt = 0; nt < 2; ++nt) {
#pragma unroll
            for (int r = 0; r < 8; ++r) {
                const int d = mWave + mt*16 + (hi ? r + 8 : r);
                const int p = pBase + nt*16 + nl;
                out[(size_t)(b*CC + d) * HW + p] = acc[mt][nt][r] + pb[d];
            }
        }
    }
}

extern "C" void kernel_launch(void* const* d_in, const int* in_sizes, int n_in,
                              void* d_out, int out_size, void* d_ws, size_t ws_size,
                              hipStream_t stream) {
    (void)in_sizes; (void)n_in; (void)out_size; (void)ws_size;
    const float* x = (const float*)d_in[0];
    const float* mw1[3] = {(const float*)d_in[1],  (const float*)d_in[7],  (const float*)d_in[13]};
    const float* mb1[3] = {(const float*)d_in[2],  (const float*)d_in[8],  (const float*)d_in[14]};
    const float* mw2[3] = {(const float*)d_in[3],  (const float*)d_in[9],  (const float*)d_in[15]};
    const float* mb2[3] = {(const float*)d_in[4],  (const float*)d_in[10], (const float*)d_in[16]};
    const float* cw[3]  = {(const float*)d_in[5],  (const float*)d_in[11], (const float*)d_in[17]};
    const float* cb[3]  = {(const float*)d_in[6],  (const float*)d_in[12], (const float*)d_in[18]};
    const float* fusion_w1 = (const float*)d_in[19];
    const float* fusion_b1 = (const float*)d_in[20];
    const float* fusion_w2 = (const float*)d_in[21];
    const float* fusion_b2 = (const float*)d_in[22];
    const float* proj_w    = (const float*)d_in[23];
    const float* proj_b    = (const float*)d_in[24];

    float* ws = (float*)d_ws;
    float* xb[3] = { ws, ws + NELEM, ws + 2*NELEM };
    float* a_buf = ws + 3*NELEM;
    float* wgt   = a_buf + BC;      // 3*BC gate weights

    for (int br = 0; br < 3; ++br) {
        mlp_kernel<<<dim3(BATCH*4, 2), 256, 0, stream>>>(
            x, mw1[br], mb1[br], mw2[br], mb2[br], xb[br]);
        dwconv_kernel<<<dim3(BATCH*4, 192), 256, 0, stream>>>(
            x, cw[br], cb[br], 2*br + 1, br, xb[br]);
    }
    mean_kernel<<<BC, 256, 0, stream>>>(xb[0], xb[1], xb[2], a_buf);
    fusion_kernel<<<BATCH, 384, 0, stream>>>(
        a_buf, fusion_w1, fusion_b1, fusion_w2, fusion_b2, wgt);
    proj_wmma_kernel<<<BATCH*32, 256, 0, stream>>>(
        xb[0], xb[1], xb[2], wgt, proj_w, proj_b, (float*)d_out);
}